// MolecularGraph_68015102099615
// MI455X (gfx1250) — compile-verified
//
#include <hip/hip_runtime.h>

// ---------------------------------------------------------------------------
// MolecularGraph GNN pipeline for gfx1250 (MI455X).
// All dense GEMMs use V_WMMA_F32_16X16X4_F32 (wave32 WMMA). The NNConv
// edge-conditioned weights are generated tile-by-tile in LDS (never hit HBM).
// Round 3: force deep unrolling in k_msg (the 6.4M-WMMA hot kernel) and hoist
// the LDS A-fragments out of the column-tile loop (invariant across tiles).
// ---------------------------------------------------------------------------

#define N_NODES   50000
#define N_EDGES   100000
#define DIM       32
#define HEADS     4
#define OUT_H     8
#define N_GRAPHS  256

typedef __attribute__((ext_vector_type(2))) float v2f;
typedef __attribute__((ext_vector_type(8))) float v8f;

// D = A(16x4) * B(4x16) + C  (fp32 WMMA, wave32)
__device__ __forceinline__ v8f wmma4(v2f a, v2f b, v8f c) {
  return __builtin_amdgcn_wmma_f32_16x16x4_f32(
      /*neg_a=*/false, a, /*neg_b=*/false, b,
      /*c_mod=*/(short)0, c, /*reuse_a=*/false, /*reuse_b=*/false);
}

__device__ __forceinline__ void atomicMaxF(float* addr, float val) {
  int* ia = (int*)addr;
  int cur = __float_as_int(*addr);
  while (__int_as_float(cur) < val) {
    int prev = atomicCAS(ia, cur, __float_as_int(val));
    if (prev == cur) break;
    cur = prev;
  }
}

// ------------------------------ utility ------------------------------------
__global__ void k_zero(float* __restrict__ p, int n, float v) {
  int i = blockIdx.x * blockDim.x + threadIdx.x;
  if (i < n) p[i] = v;
}

// nw2t[k*1024 + io] = nw2[io*32 + k]   (one-time, makes WMMA B loads coalesced)
__global__ void k_transpose_nw2(const float* __restrict__ nw2, float* __restrict__ nw2t) {
  int i = blockIdx.x * blockDim.x + threadIdx.x;
  if (i >= 32 * 1024) return;
  int io = i >> 5, k = i & 31;
  nw2t[k * 1024 + io] = nw2[i];
}

// efeat[e,k] = relu(edge_attr[e] * nw1[k] + nb1[k])
__global__ void k_edge_mlp(const float* __restrict__ ea, const float* __restrict__ nw1,
                           const float* __restrict__ nb1, float* __restrict__ efeat) {
  int i = blockIdx.x * blockDim.x + threadIdx.x;
  if (i >= N_EDGES * DIM) return;
  int e = i >> 5, k = i & 31;
  efeat[i] = fmaxf(ea[e] * nw1[k] + nb1[k], 0.0f);
}

// cnt[dst]++ over edges
__global__ void k_degree(const int* __restrict__ ei, float* __restrict__ cnt) {
  int e = blockIdx.x * blockDim.x + threadIdx.x;
  if (e >= N_EDGES) return;
  atomicAdd(&cnt[ei[N_EDGES + e]], 1.0f);
}

// ------------------- node GEMM: out = [relu](A @ Wt.T + bias) --------------
// A:[N,32], Wt:[32out,32in] (PyTorch Linear layout), out:[N,32].
// One wave per 16-node tile; two 16-col output tiles; K=32 -> 8 WMMA each.
__global__ void k_lin(const float* __restrict__ A, const float* __restrict__ Wt,
                      const float* __restrict__ bias, float* __restrict__ out,
                      int do_relu) {
  const int lane = threadIdx.x & 31;
  const int wid  = threadIdx.x >> 5;
  const int tile = blockIdx.x * 4 + wid;          // wave-uniform
  if (tile >= N_NODES / 16) return;
  const int row0 = tile * 16;
  const int l = lane & 15, hi = lane >> 4;

  v8f acc0 = {}, acc1 = {};
  #pragma unroll
  for (int k0 = 0; k0 < 32; k0 += 4) {
    v2f a, b0v, b1v;
    a.x   = A[(row0 + l) * 32 + k0 + 2 * hi];
    a.y   = A[(row0 + l) * 32 + k0 + 2 * hi + 1];
    // B[k][c] = Wt[c*32 + k]  (transposed weight; 4KB, L0-resident)
    b0v.x = Wt[(l) * 32 + k0 + 2 * hi];
    b0v.y = Wt[(l) * 32 + k0 + 2 * hi + 1];
    b1v.x = Wt[(16 + l) * 32 + k0 + 2 * hi];
    b1v.y = Wt[(16 + l) * 32 + k0 + 2 * hi + 1];
    acc0 = wmma4(a, b0v, acc0);
    acc1 = wmma4(a, b1v, acc1);
  }
  #pragma unroll
  for (int v = 0; v < 8; ++v) {
    int r = row0 + v + 8 * hi;
    float o0 = acc0[v] + (bias ? bias[l] : 0.0f);
    float o1 = acc1[v] + (bias ? bias[16 + l] : 0.0f);
    if (do_relu) { o0 = fmaxf(o0, 0.0f); o1 = fmaxf(o1, 0.0f); }
    out[r * 32 + l]      = o0;
    out[r * 32 + 16 + l] = o1;
  }
}

// ---------------- NNConv message: per-16-edge tile, W in LDS ---------------
// sW half-p holds W columns io in [p*512, p*512+512)  (i.e. input dims i in
// [p*16, p*16+16)).  WMMA computes sW = sA(16x32) @ nw2t tile-by-tile, then
// 128 threads contract msg[e,o] = sum_i h[src_e,i] * W_e[i,o], scatter-add.
#define SA_LD 33   // padded row stride: l*33 mod 64 injective -> no bank conflicts
__global__ void __launch_bounds__(128)
k_msg(const int* __restrict__ ei, const float* __restrict__ efeat,
      const float* __restrict__ h, const float* __restrict__ nw2t,
      const float* __restrict__ nb2, float* __restrict__ accum) {
  __shared__ float sA[16 * SA_LD];
  __shared__ float sH[16 * 32];
  __shared__ float sW[16 * 512];

  const int tid  = threadIdx.x;
  const int lane = tid & 31;
  const int wid  = tid >> 5;
  const int e0   = blockIdx.x * 16;

  // prefetch next tile's edge features / indices while this tile computes
  if (e0 + 16 < N_EDGES) {
    __builtin_prefetch(&efeat[(e0 + 16) * 32 + lane], 0, 1);
    __builtin_prefetch(&ei[e0 + 16], 0, 1);
  }

  for (int idx = tid; idx < 512; idx += 128) {
    int r = idx >> 5, c = idx & 31;
    sA[r * SA_LD + c] = efeat[(e0 + r) * 32 + c];
    int src = ei[e0 + r];
    sH[idx] = h[src * 32 + c];
  }
  __syncthreads();

  const int l = lane & 15, hi = lane >> 4;

  // A-fragments are invariant across all 64 column tiles: load once from LDS.
  v2f afr[8];
  #pragma unroll
  for (int kk = 0; kk < 8; ++kk) {
    afr[kk].x = sA[l * SA_LD + kk * 4 + 2 * hi];
    afr[kk].y = sA[l * SA_LD + kk * 4 + 2 * hi + 1];
  }

  float msgacc[4] = {0.f, 0.f, 0.f, 0.f};

  for (int p = 0; p < 2; ++p) {
    // --- generate 512 columns of W via WMMA (8 col-tiles per wave) ---
    #pragma unroll 4
    for (int j = 0; j < 8; ++j) {
      const int tg = p * 32 + wid * 8 + j;   // global 16-col tile, 0..63
      const int c0 = tg * 16;                // absolute io column
      v8f acc = {};
      #pragma unroll
      for (int kk = 0; kk < 8; ++kk) {
        v2f b;
        // B[k][c] = nw2t[k*1024 + c]  (coalesced: lanes 0-15 contiguous)
        b.x = nw2t[(kk * 4 + 2 * hi) * 1024 + c0 + l];
        b.y = nw2t[(kk * 4 + 2 * hi + 1) * 1024 + c0 + l];
        acc = wmma4(afr[kk], b, acc);
      }
      const int lc0 = (wid * 8 + j) * 16;    // column offset inside sW half
      #pragma unroll
      for (int v = 0; v < 8; ++v) {
        int r = v + 8 * hi;
        sW[r * 512 + lc0 + l] = acc[v] + nb2[c0 + l];
      }
    }
    __syncthreads();
    // --- partial message contraction over i in [p*16, p*16+16) ---
    #pragma unroll
    for (int j2 = 0; j2 < 4; ++j2) {
      int pair = tid + 128 * j2;             // 512 (edge,o) pairs
      int edge = pair >> 5, o = pair & 31;
      float s = 0.f;
      #pragma unroll
      for (int il = 0; il < 16; ++il)
        s += sH[edge * 32 + p * 16 + il] * sW[edge * 512 + il * 32 + o];
      msgacc[j2] += s;
    }
    __syncthreads();
  }

  #pragma unroll
  for (int j2 = 0; j2 < 4; ++j2) {
    int pair = tid + 128 * j2;
    int edge = pair >> 5, o = pair & 31;
    int dst = ei[N_EDGES + e0 + edge];
    atomicAdd(&accum[dst * 32 + o], msgacc[j2]);
  }
}

// hn = relu(accum/max(cnt,1) + h @ w_root + conv_b)   (w_root NOT transposed)
__global__ void k_nnconv_fin(const float* __restrict__ h, const float* __restrict__ accum,
                             const float* __restrict__ cnt, const float* __restrict__ w_root,
                             const float* __restrict__ conv_b, float* __restrict__ hn) {
  const int lane = threadIdx.x & 31;
  const int wid  = threadIdx.x >> 5;
  const int tile = blockIdx.x * 4 + wid;
  if (tile >= N_NODES / 16) return;
  const int row0 = tile * 16;
  const int l = lane & 15, hi = lane >> 4;

  v8f acc0 = {}, acc1 = {};
  #pragma unroll
  for (int k0 = 0; k0 < 32; k0 += 4) {
    v2f a, b0v, b1v;
    a.x   = h[(row0 + l) * 32 + k0 + 2 * hi];
    a.y   = h[(row0 + l) * 32 + k0 + 2 * hi + 1];
    // B[k][o] = w_root[k*32 + o]  (already coalesced)
    b0v.x = w_root[(k0 + 2 * hi) * 32 + l];
    b0v.y = w_root[(k0 + 2 * hi + 1) * 32 + l];
    b1v.x = w_root[(k0 + 2 * hi) * 32 + 16 + l];
    b1v.y = w_root[(k0 + 2 * hi + 1) * 32 + 16 + l];
    acc0 = wmma4(a, b0v, acc0);
    acc1 = wmma4(a, b1v, acc1);
  }
  #pragma unroll
  for (int v = 0; v < 8; ++v) {
    int r = row0 + v + 8 * hi;
    float inv = 1.0f / fmaxf(cnt[r], 1.0f);
    float o0 = accum[r * 32 + l] * inv      + acc0[v] + conv_b[l];
    float o1 = accum[r * 32 + 16 + l] * inv + acc1[v] + conv_b[16 + l];
    hn[r * 32 + l]      = fmaxf(o0, 0.0f);
    hn[r * 32 + 16 + l] = fmaxf(o1, 0.0f);
  }
}

// -------------------------------- GAT --------------------------------------
// per-node attention coefficients + init of m / denom
__global__ void k_gat_att(const float* __restrict__ z, const float* __restrict__ att_src,
                          const float* __restrict__ att_dst, float* __restrict__ a_s,
                          float* __restrict__ a_d, float* __restrict__ m,
                          float* __restrict__ denom) {
  int n = blockIdx.x * blockDim.x + threadIdx.x;
  if (n >= N_NODES) return;
  #pragma unroll
  for (int hh = 0; hh < HEADS; ++hh) {
    float s = 0.f, d = 0.f;
    #pragma unroll
    for (int f = 0; f < OUT_H; ++f) {
      float zv = z[n * 32 + hh * OUT_H + f];
      s += zv * att_src[hh * OUT_H + f];
      d += zv * att_dst[hh * OUT_H + f];
    }
    a_s[n * 4 + hh] = s;
    a_d[n * 4 + hh] = d;
    m[n * 4 + hh] = -1e30f;
    denom[n * 4 + hh] = 0.0f;
  }
}

// segment max of leaky_relu(a_s[src]+a_d[dst]) over edges + self loops
__global__ void k_gat_max(const int* __restrict__ ei, const float* __restrict__ a_s,
                          const float* __restrict__ a_d, float* __restrict__ m) {
  int id = blockIdx.x * blockDim.x + threadIdx.x;
  if (id >= N_EDGES + N_NODES) return;
  int s = (id < N_EDGES) ? ei[id] : (id - N_EDGES);
  int d = (id < N_EDGES) ? ei[N_EDGES + id] : (id - N_EDGES);
  #pragma unroll
  for (int hh = 0; hh < HEADS; ++hh) {
    float al = a_s[s * 4 + hh] + a_d[d * 4 + hh];
    al = (al > 0.f) ? al : 0.2f * al;
    atomicMaxF(&m[d * 4 + hh], al);
  }
}

// accumulate exp weights: denom[dst] += ea; num[dst,:] += ea * z[src,:]
__global__ void k_gat_accum(const int* __restrict__ ei, const float* __restrict__ a_s,
                            const float* __restrict__ a_d, const float* __restrict__ m,
                            const float* __restrict__ z, float* __restrict__ denom,
                            float* __restrict__ num) {
  int id = blockIdx.x * blockDim.x + threadIdx.x;
  if (id >= N_EDGES + N_NODES) return;
  int s = (id < N_EDGES) ? ei[id] : (id - N_EDGES);
  int d = (id < N_EDGES) ? ei[N_EDGES + id] : (id - N_EDGES);
  #pragma unroll
  for (int hh = 0; hh < HEADS; ++hh) {
    float al = a_s[s * 4 + hh] + a_d[d * 4 + hh];
    al = (al > 0.f) ? al : 0.2f * al;
    float ea = expf(al - m[d * 4 + hh]);
    atomicAdd(&denom[d * 4 + hh], ea);
    #pragma unroll
    for (int f = 0; f < OUT_H; ++f)
      atomicAdd(&num[d * 32 + hh * OUT_H + f], ea * z[s * 32 + hh * OUT_H + f]);
  }
}

// h = relu(num/denom + gat_b)
__global__ void k_gat_fin(const float* __restrict__ num, const float* __restrict__ denom,
                          const float* __restrict__ gat_b, float* __restrict__ h) {
  int i = blockIdx.x * blockDim.x + threadIdx.x;
  if (i >= N_NODES * 32) return;
  int n = i >> 5, j = i & 31;
  float v = num[i] / denom[n * 4 + (j >> 3)] + gat_b[j];
  h[i] = fmaxf(v, 0.0f);
}

// ----------------------------- pooling / head ------------------------------
__global__ void k_pool(const float* __restrict__ h, const int* __restrict__ batch,
                       float* __restrict__ gsum, float* __restrict__ gcnt) {
  int n = blockIdx.x * blockDim.x + threadIdx.x;
  if (n >= N_NODES) return;
  int b = batch[n];
  atomicAdd(&gcnt[b], 1.0f);
  for (int j = 0; j < 32; ++j)
    atomicAdd(&gsum[b * 32 + j], h[n * 32 + j]);
}

__global__ void k_head(const float* __restrict__ gsum, const float* __restrict__ gcnt,
                       const float* __restrict__ l1w, const float* __restrict__ l1b,
                       const float* __restrict__ l2w, const float* __restrict__ l2b,
                       float* __restrict__ out) {
  int b = threadIdx.x;
  if (b >= N_GRAPHS) return;
  float g[32];
  float inv = 1.0f / fmaxf(gcnt[b], 1.0f);
  #pragma unroll
  for (int j = 0; j < 32; ++j) g[j] = gsum[b * 32 + j] * inv;
  float acc = l2b[0];
  for (int k = 0; k < 16; ++k) {
    float t = l1b[k];
    #pragma unroll
    for (int j = 0; j < 32; ++j) t += g[j] * l1w[k * 32 + j];
    acc += fmaxf(t, 0.0f) * l2w[k];
  }
  out[b] = acc;
}

// ---------------------------------------------------------------------------
extern "C" void kernel_launch(void* const* d_in, const int* in_sizes, int n_in,
                              void* d_out, int out_size, void* d_ws, size_t ws_size,
                              hipStream_t stream) {
  (void)in_sizes; (void)n_in; (void)out_size; (void)ws_size;

  const float* x       = (const float*)d_in[0];
  const int*   ei      = (const int*)  d_in[1];
  const float* ea      = (const float*)d_in[2];
  const int*   batch   = (const int*)  d_in[3];
  const float* w0      = (const float*)d_in[4];
  const float* b0      = (const float*)d_in[5];
  const float* nw1     = (const float*)d_in[6];
  const float* nb1     = (const float*)d_in[7];
  const float* nw2     = (const float*)d_in[8];
  const float* nb2     = (const float*)d_in[9];
  const float* w_root  = (const float*)d_in[10];
  const float* conv_b  = (const float*)d_in[11];
  const float* gat_w   = (const float*)d_in[12];
  const float* att_src = (const float*)d_in[13];
  const float* att_dst = (const float*)d_in[14];
  const float* gat_b   = (const float*)d_in[15];
  const float* l1w     = (const float*)d_in[16];
  const float* l1b     = (const float*)d_in[17];
  const float* l2w     = (const float*)d_in[18];
  const float* l2b     = (const float*)d_in[19];
  float* out = (float*)d_out;

  // workspace carve-up (floats)
  float* p = (float*)d_ws;
  float* buf0  = p; p += N_NODES * DIM;   // h
  float* buf1  = p; p += N_NODES * DIM;   // hn (post-NNConv)
  float* bufz  = p; p += N_NODES * DIM;   // z (GAT projection)
  float* accum = p; p += N_NODES * DIM;   // NNConv mean accum / GAT numerator
  float* efeat = p; p += N_EDGES * DIM;
  float* nw2t  = p; p += 32 * 1024;       // transposed nw2 (L2-resident)
  float* cnt   = p; p += N_NODES;
  float* a_s   = p; p += N_NODES * HEADS;
  float* a_d   = p; p += N_NODES * HEADS;
  float* mmax  = p; p += N_NODES * HEADS;
  float* denom = p; p += N_NODES * HEADS;
  float* gsum  = p; p += N_GRAPHS * DIM;
  float* gcnt  = p; p += N_GRAPHS;

  const int TPB = 256;
  const int TILES     = N_NODES / 16;                  // 3125
  const int LIN_BLK   = (TILES + 3) / 4;               // 782 blocks x 4 waves
  const int MSG_BLK   = N_EDGES / 16;                  // 6250
  const int EN        = N_EDGES + N_NODES;

  // --- shared precompute (valid across both layers) ---
  k_zero<<<(N_NODES + TPB - 1) / TPB, TPB, 0, stream>>>(cnt, N_NODES, 0.0f);
  k_transpose_nw2<<<(32 * 1024 + TPB - 1) / TPB, TPB, 0, stream>>>(nw2, nw2t);
  k_edge_mlp<<<(N_EDGES * DIM + TPB - 1) / TPB, TPB, 0, stream>>>(ea, nw1, nb1, efeat);
  k_degree<<<(N_EDGES + TPB - 1) / TPB, TPB, 0, stream>>>(ei, cnt);

  // --- input linear (WMMA) ---
  k_lin<<<LIN_BLK, 128, 0, stream>>>(x, w0, b0, buf0, 1);

  for (int layer = 0; layer < 2; ++layer) {
    // NNConv
    k_zero<<<(N_NODES * DIM + TPB - 1) / TPB, TPB, 0, stream>>>(accum, N_NODES * DIM, 0.0f);
    k_msg<<<MSG_BLK, 128, 0, stream>>>(ei, efeat, buf0, nw2t, nb2, accum);
    k_nnconv_fin<<<LIN_BLK, 128, 0, stream>>>(buf0, accum, cnt, w_root, conv_b, buf1);

    // GAT
    k_lin<<<LIN_BLK, 128, 0, stream>>>(buf1, gat_w, nullptr, bufz, 0);
    k_gat_att<<<(N_NODES + TPB - 1) / TPB, TPB, 0, stream>>>(bufz, att_src, att_dst,
                                                             a_s, a_d, mmax, denom);
    k_zero<<<(N_NODES * DIM + TPB - 1) / TPB, TPB, 0, stream>>>(accum, N_NODES * DIM, 0.0f);
    k_gat_max<<<(EN + TPB - 1) / TPB, TPB, 0, stream>>>(ei, a_s, a_d, mmax);
    k_gat_accum<<<(EN + TPB - 1) / TPB, TPB, 0, stream>>>(ei, a_s, a_d, mmax, bufz, denom, accum);
    k_gat_fin<<<(N_NODES * DIM + TPB - 1) / TPB, TPB, 0, stream>>>(accum, denom, gat_b, buf0);
  }

  // --- pooling + MLP head ---
  k_zero<<<(N_GRAPHS * DIM + N_GRAPHS + TPB - 1) / TPB, TPB, 0, stream>>>(
      gsum, N_GRAPHS * DIM + N_GRAPHS, 0.0f);
  k_pool<<<(N_NODES + TPB - 1) / TPB, TPB, 0, stream>>>(buf0, batch, gsum, gcnt);
  k_head<<<1, 256, 0, stream>>>(gsum, gcnt, l1w, l1b, l2w, l2b, out);
}